// Attention_71382356459536
// MI455X (gfx1250) — compile-verified
//
#include <hip/hip_runtime.h>
#include <math.h>

#define SEQ 2048
#define NB  2048
#define DD  16
#define NWAVES 8
#define NTHREADS (NWAVES * 32)

typedef __attribute__((ext_vector_type(2))) float v2f;
typedef __attribute__((ext_vector_type(8))) float v8f;

// Fast transcendentals on the gfx1250 trans pipe, with safe fallbacks.
#if __has_builtin(__builtin_amdgcn_tanhf)
#define FAST_TANH(x) __builtin_amdgcn_tanhf(x)
#elif __has_builtin(__builtin_amdgcn_tanh_f32)
#define FAST_TANH(x) __builtin_amdgcn_tanh_f32(x)
#else
#define FAST_TANH(x) tanhf(x)
#endif

#if __has_builtin(__builtin_amdgcn_exp2f)
#define FAST_EXP2(x) __builtin_amdgcn_exp2f(x)
#else
#define FAST_EXP2(x) exp2f(x)
#endif

// exp(10*u - 10) == exp2(K*u - K), K = 10*log2(e)
#define KLOG2E 14.4269504088896340736f

// One block = 16 consecutive b-columns, all SEQ rows (softmax axis in-block).
// LDS: E[16][SEQ] unnormalized exps, qf[16][16], per-wave partials.
__global__ __launch_bounds__(NTHREADS) void
attn_ptr_net_kernel(const float* __restrict__ query,  // (1, B, D)
                    const float* __restrict__ ref,    // (SEQ, B, D)
                    const float* __restrict__ mask,   // (SEQ, B)
                    const float* __restrict__ w_q,    // (D, D)
                    const float* __restrict__ w_ref,  // (D, D)
                    const float* __restrict__ v,      // (D, 1)
                    float* __restrict__ out)          // prob (B,SEQ) ++ sample (B)
{
    extern __shared__ float smem[];
    float* Elds = smem;                  // 16*SEQ
    float* qflds = Elds + 16 * SEQ;      // 256
    float* sums = qflds + 256;           // NWAVES*16
    float* mxv  = sums + NWAVES * 16;    // NWAVES*16
    int*   mxs  = (int*)(mxv + NWAVES * 16); // NWAVES*16
    float* invs = (float*)(mxs + NWAVES * 16); // 16

    const int tid  = threadIdx.x;
    const int lane = tid & 31;
    const int wave = tid >> 5;
    const int half = lane >> 4;   // 0: lanes 0-15, 1: lanes 16-31
    const int l16  = lane & 15;
    const int b0   = blockIdx.x * 16;

    // ---- B operands for w_ref: chunk kc covers K = 4kc..4kc+3.
    // Lane layout (4x16 f32 B): VGPR0 = row (4kc+2*half), VGPR1 = row (4kc+2*half+1), col = l16.
    v2f bref[4];
#pragma unroll
    for (int kc = 0; kc < 4; ++kc) {
        int krow = 4 * kc + 2 * half;
        bref[kc][0] = w_ref[krow * DD + l16];
        bref[kc][1] = w_ref[(krow + 1) * DD + l16];
    }
    const float vlane = v[l16];

    // ---- wave 0: qf tile = query[b0:b0+16,:] @ w_q via 4 chained WMMAs.
    if (wave == 0) {
        v2f aq[4], bq[4];
#pragma unroll
        for (int kc = 0; kc < 4; ++kc) {
            int kb = 4 * kc + 2 * half;
            aq[kc] = *(const v2f*)(query + (size_t)(b0 + l16) * DD + kb);
            bq[kc][0] = w_q[kb * DD + l16];
            bq[kc][1] = w_q[(kb + 1) * DD + l16];
        }
        v8f cq = {};
#pragma unroll
        for (int kc = 0; kc < 4; ++kc)
            cq = __builtin_amdgcn_wmma_f32_16x16x4_f32(false, aq[kc], false, bq[kc],
                                                       (short)0, cq, false, false);
        // C layout: vgpr r, lane -> element (m = r + 8*half, n = l16)
#pragma unroll
        for (int r = 0; r < 8; ++r)
            qflds[(r + 8 * half) * 16 + l16] = cq[r];
    }
    __syncthreads();

    // C-init = qf tile (makes "+ query_feature" add free inside the WMMA chain)
    v8f c0;
#pragma unroll
    for (int r = 0; r < 8; ++r)
        c0[r] = qflds[(r + 8 * half) * 16 + l16];

    float sumAcc[8];
    float bestU[8];
    int   bestS[8];
#pragma unroll
    for (int r = 0; r < 8; ++r) { sumAcc[r] = 0.0f; bestU[r] = -3.4e38f; bestS[r] = 0; }

    // ---- stream over SEQ: each wave takes s = wave, wave+8, ...
    for (int s = wave; s < SEQ; s += NWAVES) {
        const float* rowp = ref + (size_t)s * NB * DD + (size_t)b0 * DD;
        v2f a[4];
#pragma unroll
        for (int kc = 0; kc < 4; ++kc)
            a[kc] = *(const v2f*)(rowp + l16 * DD + 4 * kc + 2 * half);

        // mask row for this s: lane l16 holds mask[s, b0+l16] (one load, shared via permute)
        const float mrow = mask[(size_t)s * NB + b0 + l16];

        v8f c = c0;
#pragma unroll
        for (int kc = 0; kc < 4; ++kc)
            c = __builtin_amdgcn_wmma_f32_16x16x4_f32(false, a[kc], false, bref[kc],
                                                      (short)0, c, false, false);

#pragma unroll
        for (int r = 0; r < 8; ++r) {
            // t = tanh(H[m][n]) * v[n]; reduce over n within the 16-lane half
            float t = FAST_TANH(c[r]) * vlane;
            t += __shfl_xor(t, 1);
            t += __shfl_xor(t, 2);
            t += __shfl_xor(t, 4);
            t += __shfl_xor(t, 8);
            const int m = r + 8 * half;
            float uf = FAST_TANH(t);                       // logit clip
            const float mk = __shfl(mrow, m);              // mask[s, b0+m]
            uf += (1.0f - mk) * -1e10f;
            const float e = FAST_EXP2(fmaf(uf, KLOG2E, -KLOG2E)); // exp(10u-10)
            sumAcc[r] += e;
            if (uf > bestU[r]) { bestU[r] = uf; bestS[r] = s; }
            if (l16 == 0) Elds[m * SEQ + s] = e;
        }
    }

    // ---- per-wave partials -> LDS (lanes 0 and 16 hold m=r and m=r+8)
    if (l16 == 0) {
#pragma unroll
        for (int r = 0; r < 8; ++r) {
            const int m = r + 8 * half;
            sums[wave * 16 + m] = sumAcc[r];
            mxv[wave * 16 + m]  = bestU[r];
            mxs[wave * 16 + m]  = bestS[r];
        }
    }
    __syncthreads();

    // ---- combine across waves; emit sample (argmax = first occurrence of max)
    if (tid < 16) {
        float tot = 0.0f, bv = -3.4e38f;
        int bs = 0;
#pragma unroll
        for (int w = 0; w < NWAVES; ++w) {
            tot += sums[w * 16 + tid];
            const float vv = mxv[w * 16 + tid];
            const int ss = mxs[w * 16 + tid];
            if (vv > bv || (vv == bv && ss < bs)) { bv = vv; bs = ss; }
        }
        invs[tid] = 1.0f / tot;
        out[(size_t)NB * SEQ + b0 + tid] = (float)bs;
    }
    __syncthreads();

    // ---- normalized prob, coalesced row-major (B, SEQ)
#pragma unroll 1
    for (int m = 0; m < 16; ++m) {
        const float iv = invs[m];
        float* orow = out + (size_t)(b0 + m) * SEQ;
        for (int s = tid; s < SEQ; s += NTHREADS)
            orow[s] = Elds[m * SEQ + s] * iv;
    }
}

extern "C" void kernel_launch(void* const* d_in, const int* in_sizes, int n_in,
                              void* d_out, int out_size, void* d_ws, size_t ws_size,
                              hipStream_t stream) {
    (void)in_sizes; (void)n_in; (void)d_ws; (void)ws_size; (void)out_size;
    const float* query = (const float*)d_in[0];
    const float* ref   = (const float*)d_in[1];
    const float* mask  = (const float*)d_in[2];
    const float* w_q   = (const float*)d_in[3];
    const float* w_ref = (const float*)d_in[4];
    const float* v     = (const float*)d_in[5];
    float* out = (float*)d_out;

    const size_t lds_bytes = (16 * SEQ + 256 + 3 * NWAVES * 16 + 16) * sizeof(float);
    attn_ptr_net_kernel<<<NB / 16, NTHREADS, lds_bytes, stream>>>(
        query, ref, mask, w_q, w_ref, v, out);
}